// UMAPGlobalLoss_88596585382792
// MI455X (gfx1250) — compile-verified
//
#include <hip/hip_runtime.h>
#include <hip/hip_bf16.h>

typedef __attribute__((ext_vector_type(2))) float v2f;
typedef __attribute__((ext_vector_type(8))) float v8f;

#define NROWS 4096
#define DHIGH 784
#define DEMB  16
#define LDS_STRIDE 788              // 784 padded: 788 % 64 = 20 -> conflict-free rows, 16B aligned
#define UMAP_N (4096 + 4096 * 5)    // 24576
#define A_CONST 1.576943f
#define POW_EXP 1.7901216f          // 2 * 0.8950608
#define EPS_C 1.0e-4f
#define ALPHA_C 0.9296875f
#define BETA_C 0.0703125f

// ---------------- zero workspace ----------------
__global__ __launch_bounds__(256) void zero_ws_kernel(float* __restrict__ w, int n) {
    int i = blockIdx.x * 256 + threadIdx.x;
    if (i < n) w[i] = 0.0f;
}

// ---------------- row squared norms ----------------
__global__ __launch_bounds__(256) void sq_kernel(const float* __restrict__ X,
                                                 const float* __restrict__ E,
                                                 float* __restrict__ sqh,
                                                 float* __restrict__ sql) {
    int r = blockIdx.x * 256 + threadIdx.x;
    if (r >= NROWS) return;
    const float* xr = X + (size_t)r * DHIGH;
    float s = 0.0f;
    for (int c = 0; c < DHIGH / 4; ++c) {
        float4 v = *(const float4*)(xr + 4 * c);
        s += v.x * v.x + v.y * v.y + v.z * v.z + v.w * v.w;
    }
    sqh[r] = s;
    const float* er = E + (size_t)r * 32;   // e_to = first 16 cols
    float s2 = 0.0f;
#pragma unroll
    for (int k = 0; k < DEMB; ++k) s2 += er[k] * er[k];
    sql[r] = s2;
}

// ---------------- fused WMMA gram + distance statistics ----------------
// Grid = 128 M-panels x 8 N-chunks. Each block stages a 32-row A panel
// (32 x 784 f32 = ~100 KB) in LDS; its 8 waves sweep 32 N tiles (4 each).
// Each wave computes 2 M-tiles per N-tile, so every global B fragment feeds
// two v_wmma_f32_16x16x4_f32 ops (halves L2 traffic per WMMA). Per-row
// {Σdh, Σdl, Σdh², Σdl², Σdh·dl} accumulate in registers and merge via
// global f32 atomics.
__global__ __launch_bounds__(256) void gram_stats_kernel(const float* __restrict__ X,
                                                         const float* __restrict__ E,
                                                         const float* __restrict__ sqh,
                                                         const float* __restrict__ sql,
                                                         float* __restrict__ stats) {
    __shared__ float smem[32 * LDS_STRIDE];
    const int mp = blockIdx.x >> 3;        // 0..127 : 32-row M panel
    const int chunk = blockIdx.x & 7;      // 0..7   : N chunk (32 tiles)
    const int m_base0 = mp * 32;
    const int tid = threadIdx.x;

    // cooperative fill of A panel as float4 -> ds_store_b128
    for (int i = tid; i < 32 * (DHIGH / 4); i += 256) {
        int row = i / (DHIGH / 4), c4 = i % (DHIGH / 4);
        float4 v = *(const float4*)(X + (size_t)(m_base0 + row) * DHIGH + 4 * c4);
        *(float4*)&smem[row * LDS_STRIDE + 4 * c4] = v;
    }
    __syncthreads();

    const int wave = tid >> 5;
    const int lane = tid & 31;
    const int laneN = lane & 15;    // A row / B col / C col within tile
    const int laneHi = lane >> 4;   // K split for A/B, +8 row split for C/D

    // low-dim A fragments (e_to rows) for both M tiles: K = 4*ks + 2*laneHi + {0,1}
    v2f a_low[2][4];
#pragma unroll
    for (int t = 0; t < 2; ++t) {
        const float* arow = E + (size_t)(m_base0 + t * 16 + laneN) * 32 + 2 * laneHi;
#pragma unroll
        for (int ks = 0; ks < 4; ++ks) a_low[t][ks] = *(const v2f*)(arow + 4 * ks);
    }
    float sqh_m[2][8], sql_m[2][8];
#pragma unroll
    for (int t = 0; t < 2; ++t)
#pragma unroll
        for (int r = 0; r < 8; ++r) {
            int m = m_base0 + t * 16 + r + 8 * laneHi;
            sqh_m[t][r] = sqh[m];
            sql_m[t][r] = sql[m];
        }

    float acc[5][2][8];
#pragma unroll
    for (int s = 0; s < 5; ++s)
#pragma unroll
        for (int t = 0; t < 2; ++t)
#pragma unroll
            for (int r = 0; r < 8; ++r) acc[s][t][r] = 0.0f;

    for (int it = 0; it < 4; ++it) {
        const int nt = chunk * 32 + wave + 8 * it;
        const int n_base = nt * 16;
        const float* brow_hi = X + (size_t)(n_base + laneN) * DHIGH + 2 * laneHi;
        const float* brow_lo = E + (size_t)(n_base + laneN) * 32 + 2 * laneHi;

        // prefetch next N tile's B rows into cache (global_prefetch_b8)
        if (it < 3) {
            const float* pf = X + (size_t)(n_base + 128 + laneN) * DHIGH;
            __builtin_prefetch(pf, 0, 3);
            __builtin_prefetch(pf + 196, 0, 3);
            __builtin_prefetch(pf + 392, 0, 3);
            __builtin_prefetch(pf + 588, 0, 3);
        }

        v8f ch0 = {0.f, 0.f, 0.f, 0.f, 0.f, 0.f, 0.f, 0.f};
        v8f ch1 = {0.f, 0.f, 0.f, 0.f, 0.f, 0.f, 0.f, 0.f};
        v8f cl0 = {0.f, 0.f, 0.f, 0.f, 0.f, 0.f, 0.f, 0.f};
        v8f cl1 = {0.f, 0.f, 0.f, 0.f, 0.f, 0.f, 0.f, 0.f};

        // low-dim gram: E_tile x E_tile^T  (K = 16 -> 4 steps, B reused x2)
#pragma unroll
        for (int ks = 0; ks < 4; ++ks) {
            v2f b = *(const v2f*)(brow_lo + 4 * ks);
            cl0 = __builtin_amdgcn_wmma_f32_16x16x4_f32(false, a_low[0][ks], false, b,
                                                        (short)0, cl0, false, false);
            cl1 = __builtin_amdgcn_wmma_f32_16x16x4_f32(false, a_low[1][ks], false, b,
                                                        (short)0, cl1, false, false);
        }
        // high-dim gram: X_tile x X_tile^T  (K = 784 -> 196 steps, B reused x2)
        const float* asrc0 = &smem[laneN * LDS_STRIDE + 2 * laneHi];
        const float* asrc1 = asrc0 + 16 * LDS_STRIDE;
#pragma unroll 4
        for (int ks = 0; ks < DHIGH / 4; ++ks) {
            v2f a0 = *(const v2f*)(asrc0 + 4 * ks);      // ds_load_b64
            v2f a1 = *(const v2f*)(asrc1 + 4 * ks);      // ds_load_b64
            v2f b = *(const v2f*)(brow_hi + 4 * ks);     // global_load_b64
            ch0 = __builtin_amdgcn_wmma_f32_16x16x4_f32(false, a0, false, b,
                                                        (short)0, ch0, false, false);
            ch1 = __builtin_amdgcn_wmma_f32_16x16x4_f32(false, a1, false, b,
                                                        (short)0, ch1, false, false);
        }

        const float sqh_n = sqh[n_base + laneN];
        const float sql_n = sql[n_base + laneN];
#pragma unroll
        for (int r = 0; r < 8; ++r) {
            float gh0 = ch0[r], gh1 = ch1[r], gl0 = cl0[r], gl1 = cl1[r];

            float d2h0 = fmaxf(sqh_m[0][r] + sqh_n - 2.0f * gh0, 0.0f);
            float dh0 = d2h0 > 0.0f ? sqrtf(d2h0) : 0.0f;     // _safe_sqrt
            float d2l0 = fmaxf(sql_m[0][r] + sql_n - 2.0f * gl0, 0.0f);
            float dl0 = d2l0 > 0.0f ? sqrtf(d2l0) : 0.0f;
            acc[0][0][r] += dh0;
            acc[1][0][r] += dl0;
            acc[2][0][r] += dh0 * dh0;
            acc[3][0][r] += dl0 * dl0;
            acc[4][0][r] += dh0 * dl0;

            float d2h1 = fmaxf(sqh_m[1][r] + sqh_n - 2.0f * gh1, 0.0f);
            float dh1 = d2h1 > 0.0f ? sqrtf(d2h1) : 0.0f;
            float d2l1 = fmaxf(sql_m[1][r] + sql_n - 2.0f * gl1, 0.0f);
            float dl1 = d2l1 > 0.0f ? sqrtf(d2l1) : 0.0f;
            acc[0][1][r] += dh1;
            acc[1][1][r] += dl1;
            acc[2][1][r] += dh1 * dh1;
            acc[3][1][r] += dl1 * dl1;
            acc[4][1][r] += dh1 * dl1;
        }
    }

    // reduce the 16-lane half-wave (C/D cols) and accumulate per output row
#pragma unroll
    for (int s = 0; s < 5; ++s) {
#pragma unroll
        for (int t = 0; t < 2; ++t) {
#pragma unroll
            for (int r = 0; r < 8; ++r) {
                float v = acc[s][t][r];
                v += __shfl_xor(v, 8, 32);
                v += __shfl_xor(v, 4, 32);
                v += __shfl_xor(v, 2, 32);
                v += __shfl_xor(v, 1, 32);
                if (laneN == 0) {
                    int m = m_base0 + t * 16 + r + 8 * laneHi;
                    atomicAdd(&stats[s * NROWS + m], v);
                }
            }
        }
    }
}

// ---------------- UMAP cross-entropy partial sum ----------------
__global__ __launch_bounds__(256) void umap_kernel(const float* __restrict__ E,
                                                   const int* __restrict__ perm,
                                                   float* __restrict__ accum) {
    int i = blockIdx.x * 256 + threadIdx.x;
    float term = 0.0f;
    if (i < UMAP_N) {
        const float* eto;
        const float* efrom;
        bool pos = i < NROWS;
        if (pos) {
            eto = E + (size_t)i * 32;
            efrom = eto + 16;
        } else {
            int j = i - NROWS;
            eto = E + (size_t)(j / 5) * 32;              // repeat(e_to, 5)
            efrom = E + (size_t)(perm[j] / 5) * 32 + 16; // repeat(e_from,5)[perm]
        }
        float d2 = 0.0f;
#pragma unroll
        for (int k = 0; k < DEMB; ++k) {
            float df = eto[k] - efrom[k];
            d2 += df * df;
        }
        float d = d2 > 0.0f ? sqrtf(d2) : 0.0f;
        float p = 1.0f / (1.0f + A_CONST * powf(d, POW_EXP));
        if (pos)
            term = -logf(fminf(fmaxf(p, EPS_C), 1.0f));
        else
            term = -logf(fminf(fmaxf(1.0f - p, EPS_C), 1.0f));  // REPULSION = 1
    }
    term += __shfl_xor(term, 16, 32);
    term += __shfl_xor(term, 8, 32);
    term += __shfl_xor(term, 4, 32);
    term += __shfl_xor(term, 2, 32);
    term += __shfl_xor(term, 1, 32);
    if ((threadIdx.x & 31) == 0) atomicAdd(accum, term);
}

// ---------------- finalize: Pearson corr + blend ----------------
__global__ __launch_bounds__(256) void finalize_kernel(const float* __restrict__ stats,
                                                       const float* __restrict__ umap_sum,
                                                       float* __restrict__ out) {
    __shared__ float red[256];
    const float N = (float)NROWS;
    float local = 0.0f;
    for (int r = threadIdx.x; r < NROWS; r += 256) {
        float sh = stats[0 * NROWS + r];
        float sl = stats[1 * NROWS + r];
        float sh2 = stats[2 * NROWS + r];
        float sl2 = stats[3 * NROWS + r];
        float shl = stats[4 * NROWS + r];
        float num = shl - sh * sl / N;                  // Σ (h-mh)(l-ml)
        float vh = fmaxf(sh2 - sh * sh / N, 0.0f);
        float vl = fmaxf(sl2 - sl * sl / N, 0.0f);
        float den = sqrtf(vh) * sqrtf(vl);
        local += num / den;
    }
    red[threadIdx.x] = local;
    __syncthreads();
    for (int s = 128; s > 0; s >>= 1) {
        if (threadIdx.x < s) red[threadIdx.x] += red[threadIdx.x + s];
        __syncthreads();
    }
    if (threadIdx.x == 0) {
        float loss_g = -(red[0] / N);
        float loss = umap_sum[0] / (float)UMAP_N;
        out[0] = ALPHA_C * loss + BETA_C * loss_g;
    }
}

extern "C" void kernel_launch(void* const* d_in, const int* in_sizes, int n_in,
                              void* d_out, int out_size, void* d_ws, size_t ws_size,
                              hipStream_t stream) {
    const float* E = (const float*)d_in[0];   // embed_to_from: 4096 x 32
    const float* X = (const float*)d_in[1];   // placeholder:  4096 x 784
    const int* perm = (const int*)d_in[2];    // 20480
    float* out = (float*)d_out;

    // workspace layout (floats): [0] umap_sum, [16..) sqh(4096), sql(4096), stats(5*4096)
    float* wsf = (float*)d_ws;
    float* umap_sum = wsf;
    float* sqh = wsf + 16;
    float* sql = wsf + 16 + NROWS;
    float* stats = wsf + 16 + 2 * NROWS;
    const int ws_floats = 16 + 7 * NROWS;     // 28688 floats (~112 KB)

    zero_ws_kernel<<<(ws_floats + 255) / 256, 256, 0, stream>>>(wsf, ws_floats);
    sq_kernel<<<NROWS / 256, 256, 0, stream>>>(X, E, sqh, sql);
    gram_stats_kernel<<<128 * 8, 256, 0, stream>>>(X, E, sqh, sql, stats);
    umap_kernel<<<(UMAP_N + 255) / 256, 256, 0, stream>>>(E, perm, umap_sum);
    finalize_kernel<<<1, 256, 0, stream>>>(stats, umap_sum, out);
}